// ResidualVectorQuantizer_31147102830878
// MI455X (gfx1250) — compile-verified
//
#include <hip/hip_runtime.h>
#include <stdint.h>

typedef float v4f   __attribute__((ext_vector_type(4)));
typedef int   v4i_g __attribute__((vector_size(16)));   // matches builtin param type

#define RVQ_RANK   128
#define RVQ_KRES   16
#define RVQ_BLOCK  256   // 8 waves (wave32)

// LDS holds both residual codebooks: 2 * 16 * 128 floats = 16 KB
#define LDS_CHUNKS 1024  // in float4 units: 2 * 16 * 32

__global__ __launch_bounds__(RVQ_BLOCK) void rvq_gather_kernel(
    const float* __restrict__ coarse_cb,   // [65536,128]
    const float* __restrict__ res_cb0,     // [16,128]
    const float* __restrict__ res_idx0,    // [65536,16]
    const float* __restrict__ res_cb1,     // [16,128]
    const float* __restrict__ res_idx1,    // [65536,16]
    const int*   __restrict__ coarse_idx,  // [n]
    float*       __restrict__ out,         // [n,128]
    int n)
{
    __shared__ v4f lds_cb[LDS_CHUNKS];     // [level][row 0..15][col 0..31] of float4

    const int tid = threadIdx.x;

#if __has_builtin(__builtin_amdgcn_s_wait_tensorcnt)
    __builtin_amdgcn_s_wait_tensorcnt(0);  // gfx1250 split-counter wait (no-op here)
#endif

    // ---- Stage both residual codebooks into LDS (16 KB) via async DMA -------
#if __has_builtin(__builtin_amdgcn_global_load_async_to_lds_b128)
    for (int c = tid; c < LDS_CHUNKS; c += RVQ_BLOCK) {
        const float* src = (c < 512) ? (res_cb0 + (size_t)c * 4)
                                     : (res_cb1 + (size_t)(c - 512) * 4);
        __builtin_amdgcn_global_load_async_to_lds_b128(
            (v4i_g*)(uintptr_t)src,        // global source (generic ptr, const stripped)
            (v4i_g*)&lds_cb[c],            // LDS destination (generic ptr to __shared__)
            /*offset=*/0, /*cpol=*/0);
    }
#if __has_builtin(__builtin_amdgcn_s_wait_asynccnt)
    __builtin_amdgcn_s_wait_asynccnt(0);
#else
    asm volatile("s_wait_asynccnt 0" ::: "memory");
#endif
#else
    // Fallback: plain VGPR round-trip copy
    for (int c = tid; c < LDS_CHUNKS; c += RVQ_BLOCK) {
        const float* src = (c < 512) ? (res_cb0 + (size_t)c * 4)
                                     : (res_cb1 + (size_t)(c - 512) * 4);
        lds_cb[c] = *(const v4f*)src;
    }
#endif
    __syncthreads();

    // ---- Wave-per-row main loop ---------------------------------------------
    const int lane       = tid & 31;
    const int wave       = tid >> 5;
    const int wave_gid   = blockIdx.x * (RVQ_BLOCK / 32) + wave;
    const int wave_count = gridDim.x * (RVQ_BLOCK / 32);

    const float* __restrict__ ridx_tab[2] = { res_idx0, res_idx1 };

    for (int e = wave_gid; e < n; e += wave_count) {
        // Prefetch the (sequential) index stream a few iterations ahead.
        if (lane == 0 && (e + 4 * wave_count) < n)
            __builtin_prefetch(&coarse_idx[e + 4 * wave_count], 0, 3);

        const int idx = coarse_idx[e];

        // Coalesced 512B row gather: lane handles float4 at column lane*4.
        v4f cur = *(const v4f*)(coarse_cb + (size_t)idx * RVQ_RANK + (size_t)lane * 4);

        #pragma unroll
        for (int lvl = 0; lvl < 2; ++lvl) {
            // softmax(scores) then argmax  ==  argmax(scores)  (softmax monotonic)
            int   bi = lane & 15;
            float bv = ridx_tab[lvl][(size_t)idx * RVQ_KRES + bi];

            // Wave32 argmax over 16 candidates, first-max tie-break (jnp.argmax).
            #pragma unroll
            for (int m = 8; m >= 1; m >>= 1) {
                float ov = __shfl_xor(bv, m, 16);
                int   oi = __shfl_xor(bi, m, 16);
                if (ov > bv || (ov == bv && oi < bi)) { bv = ov; bi = oi; }
            }

            // Gather-add the chosen residual row from LDS.
            cur += lds_cb[lvl * (RVQ_KRES * 32) + bi * 32 + lane];
        }

        // Output is a 512MB one-shot stream: non-temporal store (TH=NT) so it
        // does not evict the L2-resident gather tables (40MB << 192MB L2).
        __builtin_nontemporal_store(
            cur, (v4f*)(out + (size_t)e * RVQ_RANK + (size_t)lane * 4));
    }
}

extern "C" void kernel_launch(void* const* d_in, const int* in_sizes, int n_in,
                              void* d_out, int out_size, void* d_ws, size_t ws_size,
                              hipStream_t stream) {
    const float* coarse_cb = (const float*)d_in[0];
    const float* res_cb0   = (const float*)d_in[1];
    const float* res_idx0  = (const float*)d_in[2];
    const float* res_cb1   = (const float*)d_in[3];
    const float* res_idx1  = (const float*)d_in[4];
    const int*   cidx      = (const int*)d_in[5];
    float*       out       = (float*)d_out;

    const int n = in_sizes[5];              // NUM_ENTRIES
    if (n <= 0) return;

    // 2048 blocks * 8 waves = 16384 concurrent rows; each wave loops ~61 rows.
    int blocks = 2048;
    int waves_needed = (n + 7) / 8;         // if tiny problem, shrink grid
    if (blocks > waves_needed) blocks = waves_needed > 0 ? waves_needed : 1;

    rvq_gather_kernel<<<blocks, RVQ_BLOCK, 0, stream>>>(
        coarse_cb, res_cb0, res_idx0, res_cb1, res_idx1, cidx, out, n);
}